// ParametricPINN_tuned_53094385713966
// MI455X (gfx1250) — compile-verified
//
#include <hip/hip_runtime.h>

// ---------------------------------------------------------------------------
// ParametricPINN loss via forward Taylor-mode AD (bivariate jets) on gfx1250.
// Jet: f(z + t*(1,1) + s*(1,0)), coeffs c[j]: j=0..4 -> t^j, j=5..8 -> s*t^(j-5).
// All GEMMs use V_WMMA_F32_16X16X4_F32 (f32 needed for 4th-derivative fidelity).
// ---------------------------------------------------------------------------

typedef __attribute__((ext_vector_type(2))) float v2f;
typedef __attribute__((ext_vector_type(8))) float v8f;

#define TILE_M   16
#define NC       9
#define F1       512
#define F2       128
#define F3       64
#define KCHUNK   32
#define NBLK     2048          // 32768 / 16
#define NPOINTS  32768

// shared-memory float offsets (arena reused across phases)
#define OFF_H1   0                        // [NC][16][KCHUNK] = 4608 floats
#define OFF_H2   (NC * TILE_M * KCHUNK)   // [NC][16][F2]     = 18432 -> end 23040
#define OFF_H3   0                        // [NC][16][F3]     = 9216 (phase B)
#define OFF_OUTJ 9216                     // [16][NC]         = 144
#define OFF_CON  9400                     // [6][16]          = 96
#define SMEM_F   23040                    // 92160 bytes

static __device__ inline v8f zero8() {
  v8f z;
  #pragma unroll
  for (int i = 0; i < 8; ++i) z[i] = 0.f;
  return z;
}

// y = tanh(u) for jet u = p(t) + s*q(t)  (deg 4 in t, linear in s).
// Uses y' = u' * (1 - y^2); v = 1 - y^2 = sech^2 doubles as the s-channel factor.
static __device__ inline void tanh_jet(const float p[5], const float q[4],
                                       float yp[5], float yq[4]) {
  const float y0 = tanhf(p[0]);
  const float v0 = 1.f - y0 * y0;
  const float y1 = p[1] * v0;
  const float v1 = -2.f * y0 * y1;
  const float y2 = 0.5f * (p[1] * v1 + 2.f * p[2] * v0);
  const float v2 = -(2.f * y0 * y2 + y1 * y1);
  const float y3 = (1.f / 3.f) * (p[1] * v2 + 2.f * p[2] * v1 + 3.f * p[3] * v0);
  const float v3 = -2.f * (y0 * y3 + y1 * y2);
  const float y4 = 0.25f * (p[1] * v3 + 2.f * p[2] * v2 + 3.f * p[3] * v1 + 4.f * p[4] * v0);
  yp[0] = y0; yp[1] = y1; yp[2] = y2; yp[3] = y3; yp[4] = y4;
  yq[0] = q[0] * v0;
  yq[1] = q[0] * v1 + q[1] * v0;
  yq[2] = q[0] * v2 + q[1] * v1 + q[2] * v0;
  yq[3] = q[0] * v3 + q[1] * v2 + q[2] * v1 + q[3] * v0;
}

__global__ __launch_bounds__(256)
void pinn_jet_kernel(const float* __restrict__ inputs,
                     const float* __restrict__ W1, const float* __restrict__ b1,
                     const float* __restrict__ W2, const float* __restrict__ b2,
                     const float* __restrict__ W3, const float* __restrict__ b3,
                     const float* __restrict__ W4, const float* __restrict__ b4,
                     float* __restrict__ partials /* [6][3*NBLK] */) {
  __shared__ float smem[SMEM_F];
  const int tid  = threadIdx.x;
  const int lane = tid & 31;
  const int wv   = __builtin_amdgcn_readfirstlane(tid >> 5);  // wave-uniform SGPR
  const int tile = blockIdx.x;
  const int set  = blockIdx.y;        // 0: interior, 1: x=0, 2: x=L

  const int m16 = lane & 15;
  const int kb  = (lane >> 4) * 2;    // K sub-offset for f32 16x16x4 A/B frags

  // ---------------- Phase A: layer1 + tanh-jet (VALU) fused with layer2 WMMA
  const int n0 = wv * 16;             // this wave's 16 output columns of layer 2
  v8f acc[NC];
  #pragma unroll
  for (int j = 0; j < NC; ++j) acc[j] = zero8();

  for (int c = 0; c < F1 / KCHUNK; ++c) {
    __syncthreads();                  // previous chunk's GEMM reads done
    if (c + 1 < F1 / KCHUNK)
      __builtin_prefetch(&W2[(c + 1) * KCHUNK * F2], 0, 0);
    // layer-1 jets for this K-chunk of 32 units (16 pts x 32 units, 2 per thread)
    for (int t = tid; t < TILE_M * KCHUNK; t += 256) {
      const int p  = t >> 5;
      const int ul = t & 31;
      const int u  = c * KCHUNK + ul;
      const int g  = tile * TILE_M + p;
      const float a = inputs[g * 2 + 1];
      float x = inputs[g * 2 + 0];
      if (set == 1) x = 0.f;
      if (set == 2) x = 1.f;          // L = 1
      const float w0 = W1[u], w1 = W1[F1 + u];
      const float pj[5] = { x * w0 + a * w1 + b1[u], w0 + w1, 0.f, 0.f, 0.f };
      const float qj[4] = { w0, 0.f, 0.f, 0.f };
      float yp[5], yq[4];
      tanh_jet(pj, qj, yp, yq);
      #pragma unroll
      for (int i = 0; i < 5; ++i)
        smem[OFF_H1 + ((i * TILE_M + p) * KCHUNK + ul)] = yp[i];
      #pragma unroll
      for (int i = 0; i < 4; ++i)
        smem[OFF_H1 + (((5 + i) * TILE_M + p) * KCHUNK + ul)] = yq[i];
    }
    __syncthreads();
    // layer-2 GEMM over this K-chunk: 8 wmma k-steps x 9 coefficient channels
    for (int k0 = 0; k0 < KCHUNK; k0 += 4) {
      const int krow = c * KCHUNK + k0 + kb;
      v2f bfrag;
      bfrag.x = W2[(krow    ) * F2 + n0 + m16];
      bfrag.y = W2[(krow + 1) * F2 + n0 + m16];
      #pragma unroll
      for (int j = 0; j < NC; ++j) {
        v2f afrag;
        afrag.x = smem[OFF_H1 + ((j * TILE_M + m16) * KCHUNK + (k0 + kb))];
        afrag.y = smem[OFF_H1 + ((j * TILE_M + m16) * KCHUNK + (k0 + kb + 1))];
        acc[j] = __builtin_amdgcn_wmma_f32_16x16x4_f32(
            false, afrag, false, bfrag, (short)0, acc[j], false, false);
      }
    }
  }
  __syncthreads();

  // bias + tanh-jet applied IN REGISTERS: each lane slot r is one (m,n) pair
  // across all 9 coefficient accumulators; write activated jets to H2.
  {
    const float bias = b2[n0 + m16];
    #pragma unroll
    for (int r = 0; r < 8; ++r) {
      const int m = r + (lane >> 4) * 8;
      const float pj[5] = { acc[0][r] + bias, acc[1][r], acc[2][r], acc[3][r], acc[4][r] };
      const float qj[4] = { acc[5][r], acc[6][r], acc[7][r], acc[8][r] };
      float yp[5], yq[4];
      tanh_jet(pj, qj, yp, yq);
      #pragma unroll
      for (int i = 0; i < 5; ++i)
        smem[OFF_H2 + ((i * TILE_M + m) * F2 + n0 + m16)] = yp[i];
      #pragma unroll
      for (int i = 0; i < 4; ++i)
        smem[OFF_H2 + (((5 + i) * TILE_M + m) * F2 + n0 + m16)] = yq[i];
    }
  }
  __syncthreads();

  // ---------------- Phase B: layer-3 WMMA (36 = 4 tiles x 9 coeffs jobs / 8 waves)
  __builtin_prefetch(W3, 0, 0);
  v8f acc3[5];
  #pragma unroll
  for (int i = 0; i < 5; ++i) acc3[i] = zero8();
  #pragma unroll
  for (int i = 0; i < 5; ++i) {
    const int idx = wv + 8 * i;                 // wave-uniform job id
    if (idx < 4 * NC) {
      const int nt  = idx / NC;
      const int j   = idx % NC;
      const int nn0 = nt * 16;
      v8f a3 = acc3[i];
      for (int k0 = 0; k0 < F2; k0 += 4) {
        v2f afrag, bfrag;
        afrag.x = smem[OFF_H2 + ((j * TILE_M + m16) * F2 + (k0 + kb))];
        afrag.y = smem[OFF_H2 + ((j * TILE_M + m16) * F2 + (k0 + kb + 1))];
        bfrag.x = W3[(k0 + kb    ) * F3 + nn0 + m16];
        bfrag.y = W3[(k0 + kb + 1) * F3 + nn0 + m16];
        a3 = __builtin_amdgcn_wmma_f32_16x16x4_f32(
            false, afrag, false, bfrag, (short)0, a3, false, false);
      }
      acc3[i] = a3;
    }
  }
  __syncthreads();                              // H2 reads done -> safe to overlay H3
  #pragma unroll
  for (int i = 0; i < 5; ++i) {
    const int idx = wv + 8 * i;
    if (idx < 4 * NC) {
      const int nt = idx / NC, j = idx % NC, nn0 = nt * 16;
      #pragma unroll
      for (int r = 0; r < 8; ++r) {
        const int m = r + (lane >> 4) * 8;
        smem[OFF_H3 + ((j * TILE_M + m) * F3 + nn0 + m16)] = acc3[i][r];
      }
    }
  }
  __syncthreads();

  // layer-3 bias + tanh-jet per (point, unit), in place
  for (int t = tid; t < TILE_M * F3; t += 256) {
    const int p = t >> 6;
    const int u = t & 63;
    float pj[5], qj[4], yp[5], yq[4];
    pj[0] = smem[OFF_H3 + ((0 * TILE_M + p) * F3 + u)] + b3[u];
    #pragma unroll
    for (int i = 1; i < 5; ++i) pj[i] = smem[OFF_H3 + ((i * TILE_M + p) * F3 + u)];
    #pragma unroll
    for (int i = 0; i < 4; ++i) qj[i] = smem[OFF_H3 + (((5 + i) * TILE_M + p) * F3 + u)];
    tanh_jet(pj, qj, yp, yq);
    #pragma unroll
    for (int i = 0; i < 5; ++i) smem[OFF_H3 + ((i * TILE_M + p) * F3 + u)] = yp[i];
    #pragma unroll
    for (int i = 0; i < 4; ++i) smem[OFF_H3 + (((5 + i) * TILE_M + p) * F3 + u)] = yq[i];
  }
  __syncthreads();

  // ---------------- Phase C: layer 4 (64-dot per (point,coeff)) + loss terms
  if (tid < TILE_M * NC) {
    const int p = tid / NC, j = tid % NC;
    float s = (j == 0) ? b4[0] : 0.f;
    for (int u = 0; u < F3; ++u)
      s += smem[OFF_H3 + ((j * TILE_M + p) * F3 + u)] * W4[u];
    smem[OFF_OUTJ + p * NC + j] = s;
  }
  __syncthreads();

  if (tid < TILE_M) {
    const int p = tid;
    const int g = tile * TILE_M + p;
    const float a = inputs[g * 2 + 1];
    float cj[NC];
    #pragma unroll
    for (int j = 0; j < NC; ++j) cj[j] = smem[OFF_OUTJ + p * NC + j];
    const float f   = cj[0];
    const float dv1 = cj[1], dv2 = 2.f * cj[2], dv3 = 6.f * cj[3], dv4 = 24.f * cj[4];
    const float g1x = cj[5],       g1a = dv1 - g1x;
    const float g2x = cj[6],       g2a = dv2 - g2x;
    const float g3x = 2.f * cj[7], g3a = dv3 - g3x;
    const float g4x = 6.f * cj[8], g4a = dv4 - g4x;
    float term[6] = {0.f, 0.f, 0.f, 0.f, 0.f, 0.f};
    if (set == 0) {
      term[0] = (g4x + 1.f) * (g4x + 1.f) + (g4a + 1.f) * (g4a + 1.f);  // P/(E*I)=1
    } else if (set == 1) {
      term[1] = f * f;
      term[2] = g1x * g1x + g1a * g1a;
    } else {
      term[3] = (a * f) * (a * f);
      term[4] = g2x * g2x + g2a * g2a;
      const float om = 1.f - a;
      term[5] = om * om * (g3x * g3x + g3a * g3a);
    }
    #pragma unroll
    for (int q = 0; q < 6; ++q) smem[OFF_CON + q * TILE_M + p] = term[q];
  }
  __syncthreads();
  if (tid < 6) {
    float s = 0.f;
    for (int p = 0; p < TILE_M; ++p) s += smem[OFF_CON + tid * TILE_M + p];
    partials[tid * (3 * NBLK) + set * NBLK + tile] = s;   // deterministic, no atomics
  }
}

__global__ __launch_bounds__(256)
void pinn_finalize(const float* __restrict__ partials, float* __restrict__ out) {
  __shared__ float red[6][256];
  const int tid = threadIdx.x;
  float s[6] = {0.f, 0.f, 0.f, 0.f, 0.f, 0.f};
  for (int i = tid; i < 3 * NBLK; i += 256) {
    #pragma unroll
    for (int q = 0; q < 6; ++q) s[q] += partials[q * (3 * NBLK) + i];
  }
  #pragma unroll
  for (int q = 0; q < 6; ++q) red[q][tid] = s[q];
  __syncthreads();
  if (tid == 0) {
    float t[6];
    #pragma unroll
    for (int q = 0; q < 6; ++q) {
      float a = 0.f;
      for (int i = 0; i < 256; ++i) a += red[q][i];
      t[q] = a;
    }
    const float N = (float)NPOINTS;
    const float pde   = t[0] / (2.f * N);
    const float w0m   = t[1] / N;
    const float w0x   = t[2] / (2.f * N);
    const float wLm   = t[3] / N;
    const float wLxx  = t[4] / (2.f * N);
    const float wLxxx = t[5] / (2.f * N);
    out[0] = pde + w0m + w0x + wLm + wLxx + wLxxx;
    out[1] = pde;  out[2] = w0m;  out[3] = w0x;
    out[4] = wLm;  out[5] = wLxx; out[6] = wLxxx;
  }
}

extern "C" void kernel_launch(void* const* d_in, const int* in_sizes, int n_in,
                              void* d_out, int out_size, void* d_ws, size_t ws_size,
                              hipStream_t stream) {
  const float* inputs = (const float*)d_in[0];
  const float* W1 = (const float*)d_in[1];
  const float* b1 = (const float*)d_in[2];
  const float* W2 = (const float*)d_in[3];
  const float* b2 = (const float*)d_in[4];
  const float* W3 = (const float*)d_in[5];
  const float* b3 = (const float*)d_in[6];
  const float* W4 = (const float*)d_in[7];
  const float* b4 = (const float*)d_in[8];
  float* partials = (float*)d_ws;                 // 6 * 3 * 2048 floats = 147456 B

  dim3 grid(NBLK, 3);
  pinn_jet_kernel<<<grid, 256, 0, stream>>>(inputs, W1, b1, W2, b2, W3, b3, W4, b4,
                                            partials);
  pinn_finalize<<<1, 256, 0, stream>>>(partials, (float*)d_out);
}